// DifferentialGeometryOperator_86431921865222
// MI455X (gfx1250) — compile-verified
//
#include <hip/hip_runtime.h>
#include <hip/hip_bf16.h>

// ---------------------------------------------------------------------------
// DifferentialGeometryOperator for MI455X (gfx1250, wave32)
//   B=2, N=8192, D=256, K=8
//   out = [boundary_prob (B*N)] ++ [enhanced_features (B*N*D)]
// ---------------------------------------------------------------------------

typedef float v2f __attribute__((ext_vector_type(2)));
typedef float v8f __attribute__((ext_vector_type(8)));

#define KNN_K      8
#define KNN_SEG    4          // candidate-axis split for occupancy
#define D_FEAT     256
#define GRAD_SCALE 5.0f
#define ENH_SCALE  0.3f

// ---------------------------------------------------------------------------
// Kernel 1a: segmented brute-force kNN. Each thread owns one query point and
// scans ONE of 4 candidate segments (2048 candidates), keeping a register-
// resident sorted top-8. 4x the waves of a monolithic scan -> hides the
// LDS-load + compare dependency chain. Candidates staged through LDS in
// 256-point SoA tiles (broadcast reads, conflict-free).
// ---------------------------------------------------------------------------
__global__ void __launch_bounds__(256)
knn_seg_kernel(const float* __restrict__ points, float* __restrict__ cand_d,
               int* __restrict__ cand_i, int N, int segLen)
{
    __shared__ float sx[256];
    __shared__ float sy[256];
    __shared__ float sz[256];

    const int tid = threadIdx.x;
    const int seg = blockIdx.x & (KNN_SEG - 1);
    const int gid = (blockIdx.x >> 2) * 256 + tid;   // flat point id in [0, B*N)
    const int b   = gid / N;                         // uniform across block

    const float pix = points[(size_t)gid * 3 + 0];
    const float piy = points[(size_t)gid * 3 + 1];
    const float piz = points[(size_t)gid * 3 + 2];

    float bd[KNN_K];
    int   bi[KNN_K];
#pragma unroll
    for (int k = 0; k < KNN_K; ++k) { bd[k] = 3.4e38f; bi[k] = 0; }

    const float* pbase = points + (size_t)b * N * 3;
    const int t0 = seg * segLen;

    for (int t = 0; t < segLen; t += 256) {
        __syncthreads();
        const int j0 = t0 + t + tid;
        sx[tid] = pbase[(size_t)j0 * 3 + 0];
        sy[tid] = pbase[(size_t)j0 * 3 + 1];
        sz[tid] = pbase[(size_t)j0 * 3 + 2];
        __syncthreads();

        for (int jj = 0; jj < 256; ++jj) {
            const float dx = pix - sx[jj];
            const float dy = piy - sy[jj];
            const float dz = piz - sz[jj];
            const float d2 = dx * dx + dy * dy + dz * dz;
            if (d2 < bd[KNN_K - 1]) {
                bd[KNN_K - 1] = d2;
                bi[KNN_K - 1] = t0 + t + jj;
                // fully unrolled bubble-up: static indices only (no scratch)
#pragma unroll
                for (int s = KNN_K - 1; s > 0; --s) {
                    if (bd[s] < bd[s - 1]) {
                        float td = bd[s]; bd[s] = bd[s - 1]; bd[s - 1] = td;
                        int   ti = bi[s]; bi[s] = bi[s - 1]; bi[s - 1] = ti;
                    }
                }
            }
        }
    }

    const size_t base = ((size_t)gid * KNN_SEG + seg) * KNN_K;
#pragma unroll
    for (int k = 0; k < KNN_K; ++k) {
        cand_d[base + k] = bd[k];
        cand_i[base + k] = bi[k];
    }
}

// ---------------------------------------------------------------------------
// Kernel 1b: merge the 4 per-segment top-8 lists (32 candidates, contiguous
// in memory per point) into the final top-8. Segment-order insertion with
// strict '<' reproduces top_k's smallest-index tie-break.
// ---------------------------------------------------------------------------
__global__ void __launch_bounds__(256)
knn_merge_kernel(const float* __restrict__ cand_d, const int* __restrict__ cand_i,
                 int* __restrict__ knn_idx)
{
    const int gid = blockIdx.x * 256 + threadIdx.x;
    const size_t base = (size_t)gid * (KNN_SEG * KNN_K);

    float bd[KNN_K];
    int   bi[KNN_K];
#pragma unroll
    for (int k = 0; k < KNN_K; ++k) { bd[k] = 3.4e38f; bi[k] = 0; }

    for (int c = 0; c < KNN_SEG * KNN_K; ++c) {
        const float d2 = cand_d[base + c];
        const int   j  = cand_i[base + c];
        if (d2 < bd[KNN_K - 1]) {
            bd[KNN_K - 1] = d2;
            bi[KNN_K - 1] = j;
#pragma unroll
            for (int s = KNN_K - 1; s > 0; --s) {
                if (bd[s] < bd[s - 1]) {
                    float td = bd[s]; bd[s] = bd[s - 1]; bd[s - 1] = td;
                    int   ti = bi[s]; bi[s] = bi[s - 1]; bi[s - 1] = ti;
                }
            }
        }
    }

#pragma unroll
    for (int k = 0; k < KNN_K; ++k)
        knn_idx[(size_t)gid * KNN_K + k] = bi[k];
}

// ---------------------------------------------------------------------------
// Kernel 2: boundary MLP with f32 WMMA.
//   H = relu(F[32x256] * W1[256x64] + b1)   via V_WMMA_F32_16X16X4_F32
//   prob = sigmoid(H * W2[64x1] + b2)
// Block = 256 threads = 8 waves; each wave owns one 16x16 tile of H
// (2 row-tiles x 4 col-tiles). F tile staged in LDS with stride 260
// (260 % 64 == 4 -> the 16 per-row A-fragment reads land on 16 banks).
// ---------------------------------------------------------------------------
__global__ void __launch_bounds__(256)
mlp_wmma_kernel(const float* __restrict__ F, const float* __restrict__ W1,
                const float* __restrict__ b1v, const float* __restrict__ W2,
                const float* __restrict__ b2v, float* __restrict__ prob)
{
    __shared__ float sf[32 * 260];   // 32 rows x 256 cols, padded stride
    __shared__ float sh[32 * 65];    // H tile, padded stride

    const int tid     = threadIdx.x;
    const int rowBase = blockIdx.x * 32;

    // -------- stage 32x256 feature tile (float4, coalesced) --------
#pragma unroll
    for (int it = 0; it < 8; ++it) {
        const int idx4 = it * 256 + tid;       // float4 index in tile
        const int r    = idx4 >> 6;            // /64 float4 per row
        const int c    = (idx4 & 63) << 2;     // float column
        const float4 v = *(const float4*)(F + (size_t)(rowBase + r) * D_FEAT + c);
        *(float4*)(&sf[r * 260 + c]) = v;
    }
    __syncthreads();

    const int wave = tid >> 5;
    const int lane = tid & 31;
    const int rt   = wave >> 2;       // row tile: 0..1
    const int ct   = wave & 3;        // col tile: 0..3
    const int half = lane >> 4;       // 0: K=0,1   1: K=2,3   (16x4 f32 layout)
    const int lm   = lane & 15;

    const float* arow = &sf[(rt * 16 + lm) * 260];
    const float* bcol = W1 + ct * 16 + lm;

    v8f acc = {};
    for (int k = 0; k < D_FEAT; k += 4) {
        const int ka = k + half * 2;
        v2f a, bm;
        a.x  = arow[ka];
        a.y  = arow[ka + 1];
        bm.x = bcol[(size_t)ka * 64];
        bm.y = bcol[(size_t)(ka + 1) * 64];
        // D = A(16x4,f32) * B(4x16,f32) + C(16x16,f32)
        acc = __builtin_amdgcn_wmma_f32_16x16x4_f32(
            /*neg_a=*/false, a, /*neg_b=*/false, bm,
            /*c_mod=*/(short)0, acc, /*reuse_a=*/false, /*reuse_b=*/false);
    }

    // bias + ReLU, scatter 16x16 tile into LDS (C/D layout: reg r, half h ->
    // M = rt*16 + h*8 + r, N = ct*16 + lm)
    const float bias = b1v[ct * 16 + lm];
#pragma unroll
    for (int r = 0; r < 8; ++r) {
        float h = acc[r] + bias;
        h = h > 0.0f ? h : 0.0f;
        const int mrow = rt * 16 + half * 8 + r;
        sh[mrow * 65 + ct * 16 + lm] = h;
    }
    __syncthreads();

    // second layer: 64-wide dot + sigmoid, one thread per row
    if (tid < 32) {
        float s = 0.0f;
#pragma unroll
        for (int n = 0; n < 64; ++n)
            s += sh[tid * 65 + n] * W2[n];
        s += b2v[0];
        prob[rowBase + tid] = 1.0f / (1.0f + __expf(-s));
    }
}

// ---------------------------------------------------------------------------
// Kernel 3: neighbor-feature gradient + enhancement.
// One wave per point; 8 consecutive floats per lane (2x float4 = b128 loads,
// each neighbor row = 32 lanes x 32B = 1KB fully coalesced).
// Butterfly shfl_xor reduction across the wave32.
// ---------------------------------------------------------------------------
__global__ void __launch_bounds__(256)
grad_enhance_kernel(const float* __restrict__ F, const int* __restrict__ knn_idx,
                    const float* __restrict__ prob, float* __restrict__ outF, int N)
{
    const int tid  = threadIdx.x;
    const int wave = tid >> 5;
    const int lane = tid & 31;
    const size_t p = (size_t)blockIdx.x * 8 + wave;   // flat point id
    const int    b = (int)(p / N);

    const float* fc = F + p * D_FEAT + lane * 8;
    const float4 c0 = *(const float4*)(fc);
    const float4 c1 = *(const float4*)(fc + 4);

    const float* fbase = F + (size_t)b * N * D_FEAT;

    float gsum = 0.0f;
#pragma unroll
    for (int k = 0; k < KNN_K; ++k) {
        const int j = knn_idx[p * KNN_K + k];           // broadcast load
        const float* fn = fbase + (size_t)j * D_FEAT + lane * 8;
        const float4 n0 = *(const float4*)(fn);
        const float4 n1 = *(const float4*)(fn + 4);

        float d, s;
        d = n0.x - c0.x; s  = d * d;
        d = n0.y - c0.y; s += d * d;
        d = n0.z - c0.z; s += d * d;
        d = n0.w - c0.w; s += d * d;
        d = n1.x - c1.x; s += d * d;
        d = n1.y - c1.y; s += d * d;
        d = n1.z - c1.z; s += d * d;
        d = n1.w - c1.w; s += d * d;

#pragma unroll
        for (int off = 16; off >= 1; off >>= 1)
            s += __shfl_xor(s, off, 32);
        gsum += sqrtf(s);
    }

    const float grad  = gsum * (1.0f / KNN_K);
    const float scale = ENH_SCALE * tanhf(GRAD_SCALE * grad) * prob[p];

    float* fo = outF + p * D_FEAT + lane * 8;
    float4 o0 = c0, o1 = c1;
    o0.x += scale; o0.y += scale; o0.z += scale; o0.w += scale;
    o1.x += scale; o1.y += scale; o1.z += scale; o1.w += scale;
    *(float4*)(fo)     = o0;
    *(float4*)(fo + 4) = o1;
}

// ---------------------------------------------------------------------------
// Launch
// ---------------------------------------------------------------------------
extern "C" void kernel_launch(void* const* d_in, const int* in_sizes, int n_in,
                              void* d_out, int out_size, void* d_ws, size_t ws_size,
                              hipStream_t stream)
{
    const float* features = (const float*)d_in[0];   // [B,N,D]
    const float* points   = (const float*)d_in[1];   // [B,N,3]
    const float* W1       = (const float*)d_in[2];   // [D,64]
    const float* b1       = (const float*)d_in[3];   // [64]
    const float* W2       = (const float*)d_in[4];   // [64,1]
    const float* b2       = (const float*)d_in[5];   // [1]

    const int N  = 8192;
    const int BN = in_sizes[1] / 3;                  // B*N = 16384

    float* prob = (float*)d_out;                     // [B*N]
    float* outF = (float*)d_out + BN;                // [B*N, D]

    // workspace layout
    float* cand_d = (float*)d_ws;                            // BN*4*8 floats
    int*   cand_i = (int*)(cand_d + (size_t)BN * KNN_SEG * KNN_K);
    int*   knn    = (int*)(cand_i + (size_t)BN * KNN_SEG * KNN_K);

    // 1) segmented kNN (4x occupancy) + merge
    knn_seg_kernel<<<(BN / 256) * KNN_SEG, 256, 0, stream>>>(points, cand_d, cand_i,
                                                             N, N / KNN_SEG);
    knn_merge_kernel<<<BN / 256, 256, 0, stream>>>(cand_d, cand_i, knn);

    // 2) boundary probability via f32 WMMA
    mlp_wmma_kernel<<<BN / 32, 256, 0, stream>>>(features, W1, b1, W2, b2, prob);

    // 3) feature gradient + enhancement
    grad_enhance_kernel<<<BN / 8, 256, 0, stream>>>(features, knn, prob, outF, N);
}